// HardwareOptimizedQuantumLLM_34187939676379
// MI455X (gfx1250) — compile-verified
//
#include <hip/hip_runtime.h>
#include <math.h>

#define Bq 4
#define Sq 1024
#define Dq 512
#define Vq 8192
#define NSC 5
#define WINR 8
#define WINW 17
#define EPSq 1e-5f

// GEMM tiling
#define BM 128
#define BN 64
#define BKq 16

typedef __attribute__((ext_vector_type(2))) float v2f;
typedef __attribute__((ext_vector_type(8))) float v8f;

__device__ __forceinline__ float gelu_exact(float x) {
    return 0.5f * x * (1.0f + erff(x * 0.70710678118654752f));
}

// ---------------- Stage 1: phases + cumulative sums (one thread per (b,d) column)
__global__ void k_cumsum(const float* __restrict__ emb, float* __restrict__ phases,
                         float* __restrict__ cumP, float* __restrict__ cumC,
                         float* __restrict__ cumS) {
    int t = blockIdx.x * blockDim.x + threadIdx.x;
    if (t >= Bq * Dq) return;
    int b = t / Dq, d = t % Dq;
    float sp = 0.f, sc = 0.f, ss = 0.f;
    size_t cbase = (size_t)b * (Sq + 1) * Dq + d;
    cumP[cbase] = 0.f; cumC[cbase] = 0.f; cumS[cbase] = 0.f;
    for (int i = 0; i < Sq; i++) {
        float p = tanhf(emb[((size_t)(b * Sq + i)) * Dq + d]);
        phases[((size_t)(b * Sq + i)) * Dq + d] = p;
        sp += p; sc += cosf(p); ss += sinf(p);
        size_t ci = cbase + (size_t)(i + 1) * Dq;
        cumP[ci] = sp; cumC[ci] = sc; cumS[ci] = ss;
    }
}

// ---------------- Stage 2: multi-scale coherence means -> msc[b,s,5]
__global__ void k_msc(const float* __restrict__ cumP, const float* __restrict__ cumC,
                      const float* __restrict__ cumS, float* __restrict__ msc) {
    __shared__ float red[256];
    int bid = blockIdx.x;
    int b = bid / Sq, i = bid % Sq;
    const int scales[NSC] = {1, 2, 4, 8, 16};
    float acc[NSC];
#pragma unroll
    for (int k = 0; k < NSC; k++) acc[k] = 0.f;
    size_t base = (size_t)b * (Sq + 1) * Dq;
    for (int d = threadIdx.x; d < Dq; d += blockDim.x) {
#pragma unroll
        for (int k = 0; k < NSC; k++) {
            int s = scales[k];
            if (i + s <= Sq) {
                float inv = 1.0f / (float)s;
                size_t hi = base + (size_t)(i + s) * Dq + d;
                size_t lo = base + (size_t)i * Dq + d;
                float wm = (cumP[hi] - cumP[lo]) * inv;
                float ac = (cumC[hi] - cumC[lo]) * inv;
                float as = (cumS[hi] - cumS[lo]) * inv;
                acc[k] += cosf(wm) * ac + sinf(wm) * as;
            }
        }
    }
    for (int k = 0; k < NSC; k++) {
        red[threadIdx.x] = acc[k];
        __syncthreads();
        for (int off = 128; off > 0; off >>= 1) {
            if (threadIdx.x < off) red[threadIdx.x] += red[threadIdx.x + off];
            __syncthreads();
        }
        if (threadIdx.x == 0) msc[(size_t)bid * NSC + k] = red[0] * (1.0f / Dq);
        __syncthreads();
    }
}

// ---------------- Stage 3: inverse row norms of gathered token_similarity rows
__global__ void k_norm(const int* __restrict__ ids, const float* __restrict__ tsim,
                       float* __restrict__ invn) {
    __shared__ float red[256];
    int bi = blockIdx.x;
    const float* row = tsim + (size_t)ids[bi] * Vq;
    float ss = 0.f;
    for (int v = threadIdx.x; v < Vq; v += blockDim.x) { float x = row[v]; ss += x * x; }
    red[threadIdx.x] = ss;
    __syncthreads();
    for (int off = 128; off > 0; off >>= 1) {
        if (threadIdx.x < off) red[threadIdx.x] += red[threadIdx.x + off];
        __syncthreads();
    }
    if (threadIdx.x == 0) invn[bi] = 1.0f / fmaxf(sqrtf(red[0]), 1e-12f);
}

// ---------------- Stage 4: banded similarity swin[b,i,w] (only the 17-wide band is needed)
__global__ void k_swin(const int* __restrict__ ids, const float* __restrict__ tsim,
                       const float* __restrict__ invn, float* __restrict__ swin) {
    __shared__ float rowi[Vq];   // 32 KB: cache row i, stream rows j (L2-resident)
    __shared__ float red[256];
    int bi = blockIdx.x;
    int b = bi / Sq, i = bi % Sq;
    const float* ri = tsim + (size_t)ids[bi] * Vq;
    for (int v = threadIdx.x; v < Vq; v += blockDim.x) rowi[v] = ri[v];
    __syncthreads();
    for (int w = 0; w < WINW; w++) {
        int j = i - WINR + w;
        float out = 0.f;
        if (j >= 0 && j < Sq) {   // uniform across block
            const float* rj = tsim + (size_t)ids[b * Sq + j] * Vq;
            float dot = 0.f;
            for (int v = threadIdx.x; v < Vq; v += blockDim.x) dot += rowi[v] * rj[v];
            red[threadIdx.x] = dot;
            __syncthreads();
            for (int off = 128; off > 0; off >>= 1) {
                if (threadIdx.x < off) red[threadIdx.x] += red[threadIdx.x + off];
                __syncthreads();
            }
            out = red[0] * invn[bi] * invn[b * Sq + j];
            __syncthreads();
        }
        if (threadIdx.x == 0) swin[(size_t)bi * WINW + w] = out;
    }
}

// ---------------- Stage 5: semantic coherence sem[i]
__global__ void k_sem(const float* __restrict__ phases, const float* __restrict__ swin,
                      float* __restrict__ sem) {
    __shared__ float red[256];
    int i = blockIdx.x;
    float acc = 0.f;
    const int total = Bq * WINW * Dq;
    for (int t = threadIdx.x; t < total; t += blockDim.x) {
        int d = t % Dq;
        int rest = t / Dq;
        int w = rest % WINW;
        int b = rest / WINW;
        int j = i - WINR + w;
        if (j < 0 || j >= Sq) continue;
        float pi = phases[((size_t)(b * Sq + i)) * Dq + d];
        float pj = phases[((size_t)(b * Sq + j)) * Dq + d];
        float sw = swin[((size_t)(b * Sq + i)) * WINW + w];
        acc += cosf((pi - pj) * sw);
    }
    red[threadIdx.x] = acc;
    __syncthreads();
    for (int off = 128; off > 0; off >>= 1) {
        if (threadIdx.x < off) red[threadIdx.x] += red[threadIdx.x + off];
        __syncthreads();
    }
    if (threadIdx.x == 0) {
        int cnt = min(i, WINR) + min(Sq - 1 - i, WINR) + 1;
        sem[i] = red[0] / ((float)Bq * (float)cnt * (float)Dq);
    }
}

// ---------------- Stage 6: context mean + cos(e - cmean)
__global__ void k_cmean(const float* __restrict__ emb, const float* __restrict__ mask,
                        float* __restrict__ cmean) {
    int t = blockIdx.x * blockDim.x + threadIdx.x;
    if (t >= Bq * Dq) return;
    int b = t / Dq, d = t % Dq;
    float s = 0.f;
    for (int i = 0; i < Sq; i++) s += emb[((size_t)(b * Sq + i)) * Dq + d] * mask[b * Sq + i];
    cmean[t] = s / (float)Sq;
}

__global__ void k_ctxin(const float* __restrict__ emb, const float* __restrict__ cmean,
                        float* __restrict__ ctx) {
    size_t t = (size_t)blockIdx.x * blockDim.x + threadIdx.x;
    if (t >= (size_t)Bq * Sq * Dq) return;
    int d = (int)(t % Dq);
    int bs = (int)(t / Dq);
    int b = bs / Sq;
    ctx[t] = cosf(emb[t] - cmean[b * Dq + d]);
}

// ---------------- WMMA f32 GEMM:  C[M,N] = A[M,K] @ W[K,N] + bias[N]
// block = 256 threads = 8 waves (4 along M x 2 along N) -> 128x64 block tile,
// each wave owns a 2x2 grid of 16x16 f32 accumulator tiles (32x32).
// K staged through LDS in chunks of 16: As is K-major (transposed on store) so
// fragment reads across lanes 0-15 are bank-conflict-free; Bs is N-major.
__global__ void k_gemm_wmma(const float* __restrict__ A, int lda,
                            const float* __restrict__ W, int N,
                            const float* __restrict__ bias,
                            float* __restrict__ C, int ldc, int K) {
    __shared__ float As[BKq][BM];   // 8 KB
    __shared__ float Bs[BKq][BN];   // 4 KB
    const int tid   = threadIdx.x;
    const int lane  = tid & 31;
    const int waveM = (tid >> 5) >> 1;   // 0..3
    const int waveN = (tid >> 5) & 1;    // 0..1
    const int l16   = lane & 15;
    const int khalf = (lane >> 4) << 1;  // lanes 0-15: K+0/K+1 ; lanes 16-31: K+2/K+3
    const int blockM = blockIdx.x * BM;
    const int blockN = blockIdx.y * BN;

    // staging assignments: A tile 128x16 (8 elems/thread), B tile 16x64 (4 elems/thread)
    const int arow = tid >> 1;            // 0..127
    const int ak8  = (tid & 1) * 8;       // 0 or 8
    const int brow = tid >> 4;            // 0..15
    const int bn4  = (tid & 15) * 4;      // 0..60

    v8f acc[2][2];
#pragma unroll
    for (int mi = 0; mi < 2; mi++)
#pragma unroll
        for (int ni = 0; ni < 2; ni++)
#pragma unroll
            for (int r = 0; r < 8; r++) acc[mi][ni][r] = 0.f;

    for (int kb = 0; kb < K; kb += BKq) {
        __syncthreads();
        // stage A (transpose to K-major), 2x float4 coalesced loads
        const float* agp = A + (size_t)(blockM + arow) * lda + kb + ak8;
        float4 a0 = ((const float4*)agp)[0];
        float4 a1 = ((const float4*)agp)[1];
        As[ak8 + 0][arow] = a0.x; As[ak8 + 1][arow] = a0.y;
        As[ak8 + 2][arow] = a0.z; As[ak8 + 3][arow] = a0.w;
        As[ak8 + 4][arow] = a1.x; As[ak8 + 5][arow] = a1.y;
        As[ak8 + 6][arow] = a1.z; As[ak8 + 7][arow] = a1.w;
        // stage B, one float4 coalesced load
        float4 bv4 = *(const float4*)(W + (size_t)(kb + brow) * N + blockN + bn4);
        *(float4*)&Bs[brow][bn4] = bv4;
        // prefetch next A stage into cache (global_prefetch_b8)
        if (kb + BKq < K)
            __builtin_prefetch(A + (size_t)(blockM + arow) * lda + kb + BKq + ak8, 0, 1);
        __syncthreads();

#pragma unroll
        for (int kk = 0; kk < BKq; kk += 4) {
            v2f af[2], bf[2];
#pragma unroll
            for (int mi = 0; mi < 2; mi++) {
                int m = waveM * 32 + mi * 16 + l16;
                af[mi].x = As[kk + khalf][m];
                af[mi].y = As[kk + khalf + 1][m];
            }
#pragma unroll
            for (int ni = 0; ni < 2; ni++) {
                int n = waveN * 32 + ni * 16 + l16;
                bf[ni].x = Bs[kk + khalf][n];
                bf[ni].y = Bs[kk + khalf + 1][n];
            }
#pragma unroll
            for (int mi = 0; mi < 2; mi++)
#pragma unroll
                for (int ni = 0; ni < 2; ni++)
                    acc[mi][ni] = __builtin_amdgcn_wmma_f32_16x16x4_f32(
                        /*neg_a=*/false, af[mi], /*neg_b=*/false, bf[ni],
                        /*c_mod=*/(short)0, acc[mi][ni],
                        /*reuse_a=*/false, /*reuse_b=*/false);
        }
    }

    // epilogue: C layout -> VGPR r holds rows r (lanes 0-15) / r+8 (lanes 16-31)
#pragma unroll
    for (int mi = 0; mi < 2; mi++) {
        int srow = blockM + waveM * 32 + mi * 16 + (lane >> 4) * 8;
#pragma unroll
        for (int ni = 0; ni < 2; ni++) {
            int col = blockN + waveN * 32 + ni * 16 + l16;
            float bvv = bias[col];
#pragma unroll
            for (int r = 0; r < 8; r++)
                C[(size_t)(srow + r) * ldc + col] = acc[mi][ni][r] + bvv;
        }
    }
}

// ---------------- Fill allc columns [0,D): scale_coh, [D,2D): sem_coh, [3D,4D): local coherence
__global__ void k_allc_fill(const float* __restrict__ emb, const float* __restrict__ msc,
                            const float* __restrict__ sem, const float* __restrict__ Wsc,
                            const float* __restrict__ bsc, const float* __restrict__ Wsem,
                            const float* __restrict__ bsem, float* __restrict__ allc) {
    size_t t = (size_t)blockIdx.x * blockDim.x + threadIdx.x;
    if (t >= (size_t)Bq * Sq * Dq) return;
    int d = (int)(t % Dq);
    int bs = (int)(t / Dq);
    int s = bs % Sq;
    float sc = bsc[d];
#pragma unroll
    for (int k = 0; k < NSC; k++) sc += msc[(size_t)bs * NSC + k] * Wsc[k * Dq + d];
    float se = sem[s] * Wsem[d] + bsem[d];
    float lc = 0.f;
    if (s < Sq - 1) lc = cosf(emb[t + Dq] - emb[t]);
    size_t rb = (size_t)bs * (4 * Dq);
    allc[rb + d] = sc;
    allc[rb + Dq + d] = se;
    allc[rb + 3 * Dq + d] = lc;
}

// ---------------- LayerNorm over D=512 (block=256, 2 elems/thread), optional exact GELU
__global__ void k_layernorm(const float* __restrict__ in, float* __restrict__ out,
                            const float* __restrict__ g, const float* __restrict__ bb,
                            int do_gelu) {
    __shared__ float red[256];
    int row = blockIdx.x;
    const float* x = in + (size_t)row * Dq;
    float v0 = x[threadIdx.x];
    float v1 = x[threadIdx.x + 256];
    red[threadIdx.x] = v0 + v1;
    __syncthreads();
    for (int off = 128; off > 0; off >>= 1) {
        if (threadIdx.x < off) red[threadIdx.x] += red[threadIdx.x + off];
        __syncthreads();
    }
    float m = red[0] * (1.0f / Dq);
    __syncthreads();
    float d0 = v0 - m, d1 = v1 - m;
    red[threadIdx.x] = d0 * d0 + d1 * d1;
    __syncthreads();
    for (int off = 128; off > 0; off >>= 1) {
        if (threadIdx.x < off) red[threadIdx.x] += red[threadIdx.x + off];
        __syncthreads();
    }
    float rstd = rsqrtf(red[0] * (1.0f / Dq) + EPSq);
    float o0 = d0 * rstd * g[threadIdx.x] + bb[threadIdx.x];
    float o1 = d1 * rstd * g[threadIdx.x + 256] + bb[threadIdx.x + 256];
    if (do_gelu) { o0 = gelu_exact(o0); o1 = gelu_exact(o1); }
    out[(size_t)row * Dq + threadIdx.x] = o0;
    out[(size_t)row * Dq + threadIdx.x + 256] = o1;
}

// ---------------- Residual: x += 0.1 * gelu(h)
__global__ void k_resid(float* __restrict__ x, const float* __restrict__ h) {
    size_t t = (size_t)blockIdx.x * blockDim.x + threadIdx.x;
    if (t >= (size_t)Bq * Sq * Dq) return;
    x[t] += 0.1f * gelu_exact(h[t]);
}

extern "C" void kernel_launch(void* const* d_in, const int* in_sizes, int n_in,
                              void* d_out, int out_size, void* d_ws, size_t ws_size,
                              hipStream_t stream) {
    (void)in_sizes; (void)n_in; (void)out_size; (void)ws_size;

    const float* emb   = (const float*)d_in[0];
    const int*   ids   = (const int*)d_in[1];
    const float* mask  = (const float*)d_in[2];
    const float* tsim  = (const float*)d_in[3];
    const float* Wsc   = (const float*)d_in[4];
    const float* bsc   = (const float*)d_in[5];
    const float* Wsem  = (const float*)d_in[6];
    const float* bsem  = (const float*)d_in[7];
    const float* Wctx  = (const float*)d_in[8];
    const float* bctx  = (const float*)d_in[9];
    const float* Wfuse = (const float*)d_in[10];
    const float* bfuse = (const float*)d_in[11];
    const float* ln1g  = (const float*)d_in[12];
    const float* ln1b  = (const float*)d_in[13];
    const float* ln2g  = (const float*)d_in[14];
    const float* ln2b  = (const float*)d_in[15];
    const float* ln3g  = (const float*)d_in[16];
    const float* ln3b  = (const float*)d_in[17];
    const float* We[3] = {(const float*)d_in[18], (const float*)d_in[20], (const float*)d_in[22]};
    const float* be[3] = {(const float*)d_in[19], (const float*)d_in[21], (const float*)d_in[23]};

    float* ws = (float*)d_ws;
    const size_t NPH  = (size_t)Bq * Sq * Dq;        // 2,097,152
    const size_t CUM  = (size_t)Bq * (Sq + 1) * Dq;  // 2,099,200
    const size_t NALL = (size_t)Bq * Sq * 4 * Dq;    // 8,388,608

    float* allc   = ws;                 // 32 MB arena (early: cumsum arrays; later: allc)
    float* cumP   = allc;
    float* cumC   = allc + CUM;
    float* cumS   = allc + 2 * CUM;
    float* phases = ws + NALL;
    float* xbuf   = phases + NPH;
    float* tbuf   = xbuf + NPH;
    float* hbuf   = tbuf + NPH;         // also ctx_in (used before residual loop)
    float* ctxin  = hbuf;
    float* smallb = hbuf + NPH;
    float* msc    = smallb;                               // B*S*5
    float* invn   = msc + (size_t)Bq * Sq * NSC;          // B*S
    float* swin   = invn + (size_t)Bq * Sq;               // B*S*17
    float* sem    = swin + (size_t)Bq * Sq * WINW;        // S
    float* cmean  = sem + Sq;                             // B*D

    const dim3 gemmGrid(Bq * Sq / BM, Dq / BN);           // (32, 8)

    // Stage 1-2: phases, cumsums, multi-scale coherence
    k_cumsum<<<(Bq * Dq + 255) / 256, 256, 0, stream>>>(emb, phases, cumP, cumC, cumS);
    k_msc<<<Bq * Sq, 256, 0, stream>>>(cumP, cumC, cumS, msc);

    // Stage 3-5: banded token similarity + semantic coherence
    k_norm<<<Bq * Sq, 256, 0, stream>>>(ids, tsim, invn);
    k_swin<<<Bq * Sq, 256, 0, stream>>>(ids, tsim, invn, swin);
    k_sem<<<Sq, 256, 0, stream>>>(phases, swin, sem);

    // Stage 6: context coherence input
    k_cmean<<<(Bq * Dq + 255) / 256, 256, 0, stream>>>(emb, mask, cmean);
    k_ctxin<<<(int)(NPH / 256), 256, 0, stream>>>(emb, cmean, ctxin);

    // ctx GEMM into allc columns [2D,3D) (cumsum arena is dead now)
    k_gemm_wmma<<<gemmGrid, 256, 0, stream>>>(ctxin, Dq, Wctx, Dq, bctx,
                                              allc + 2 * Dq, 4 * Dq, Dq);
    // remaining allc columns
    k_allc_fill<<<(int)(NPH / 256), 256, 0, stream>>>(emb, msc, sem, Wsc, bsc, Wsem, bsem, allc);

    // fuse GEMM (K = 2048) -> x, then LN1 + GELU in-place
    k_gemm_wmma<<<gemmGrid, 256, 0, stream>>>(allc, 4 * Dq, Wfuse, Dq, bfuse,
                                              xbuf, Dq, 4 * Dq);
    k_layernorm<<<Bq * Sq, 256, 0, stream>>>(xbuf, xbuf, ln1g, ln1b, 1);

    // 3 residual blocks: t = LN2(x); h = t @ W_e + b_e; x += 0.1*gelu(h)
    for (int l = 0; l < 3; l++) {
        k_layernorm<<<Bq * Sq, 256, 0, stream>>>(xbuf, tbuf, ln2g, ln2b, 0);
        k_gemm_wmma<<<gemmGrid, 256, 0, stream>>>(tbuf, Dq, We[l], Dq, be[l],
                                                  hbuf, Dq, Dq);
        k_resid<<<(int)(NPH / 256), 256, 0, stream>>>(xbuf, hbuf);
    }

    // final LN -> output
    k_layernorm<<<Bq * Sq, 256, 0, stream>>>(xbuf, (float*)d_out, ln3g, ln3b, 0);
}